// GatedEMA_8521215115370
// MI455X (gfx1250) — compile-verified
//
#include <hip/hip_runtime.h>
#include <cstdint>

// Problem constants (match reference)
#define BB 4
#define SS 8192
#define DD 1024
#define NDIM 3
#define SCALE_F 0.5773502691896258f   // (1/3)^0.5

// Scan decomposition
#define CHUNKS 64
#define CLEN   (SS / CHUNKS)          // 128 steps per chunk
#define NT     256                    // threads per block (8 waves)
#define VEC    4                      // channels per thread (float4 path)
#define STEP   8                      // sequence rows per LDS tile
#define TILES  (CLEN / STEP)          // 16 tiles per chunk

#ifndef __has_builtin
#define __has_builtin(x) 0
#endif

#if defined(__HIP_DEVICE_COMPILE__) &&                                \
    __has_builtin(__builtin_amdgcn_global_load_async_to_lds_b128) &&  \
    __has_builtin(__builtin_amdgcn_s_wait_asynccnt)
#define USE_ASYNC_LDS 1
#else
#define USE_ASYNC_LDS 0
#endif

#if USE_ASYNC_LDS
typedef int v4i __attribute__((ext_vector_type(4)));
typedef __attribute__((address_space(1))) v4i* g_v4i_ptr;  // global v4i*
typedef __attribute__((address_space(3))) v4i* l_v4i_ptr;  // LDS v4i*
#endif

__device__ __forceinline__ float sigmoid_f(float v) {
    return 1.0f / (1.0f + __expf(-v));
}

// EMA parameters for VEC consecutive channels starting at d0:
// q[v][n] decay, w[v][n] output weight (w may be skipped by passing nullptr
// semantics via template-free overload below).
__device__ __forceinline__ void load_q(const float* __restrict__ delta,
                                       const float* __restrict__ alpha,
                                       int d0, float q[VEC][NDIM]) {
#pragma unroll
    for (int v = 0; v < VEC; ++v) {
#pragma unroll
        for (int n = 0; n < NDIM; ++n) {
            float p = sigmoid_f(delta[(d0 + v) * NDIM + n]);
            q[v][n] = 1.0f - p * sigmoid_f(alpha[(d0 + v) * NDIM + n]);
        }
    }
}

__device__ __forceinline__ void load_qw(const float* __restrict__ delta,
                                        const float* __restrict__ alpha,
                                        const float* __restrict__ beta,
                                        const float* __restrict__ gamma,
                                        int d0, float q[VEC][NDIM],
                                        float w[VEC][NDIM]) {
#pragma unroll
    for (int v = 0; v < VEC; ++v) {
#pragma unroll
        for (int n = 0; n < NDIM; ++n) {
            float p = sigmoid_f(delta[(d0 + v) * NDIM + n]);
            q[v][n] = 1.0f - p * sigmoid_f(alpha[(d0 + v) * NDIM + n]);
            w[v][n] = p * beta[(d0 + v) * NDIM + n] *
                      gamma[(d0 + v) * NDIM + n] * SCALE_F;
        }
    }
}

// ---------------------------------------------------------------------------
// Kernel 1: chunk-local scan (zero seed) -> chunk-end states E.
// float4 per thread: block covers all D=1024 channels; 512B/wave transactions.
// Also streams all of x once, leaving it resident in the 192MB L2.
// carry layout: [(b*DD + d)*NDIM + n][CHUNKS]
// ---------------------------------------------------------------------------
__global__ void __launch_bounds__(NT)
k1_chunk_end(const float* __restrict__ x, const float* __restrict__ delta,
             const float* __restrict__ alpha, float* __restrict__ carry) {
    const int c   = blockIdx.x;
    const int b   = blockIdx.z;
    const int tid = threadIdx.x;
    const int d0  = tid * VEC;

    float q[VEC][NDIM];
    load_q(delta, alpha, d0, q);

    float h[VEC][NDIM];
#pragma unroll
    for (int v = 0; v < VEC; ++v)
#pragma unroll
        for (int n = 0; n < NDIM; ++n) h[v][n] = 0.0f;

    const float4* xp4 =
        (const float4*)(x + ((size_t)b * SS + (size_t)c * CLEN) * DD) + tid;
    const int rowstride4 = DD / 4;  // float4 elements per sequence row

    for (int sg = 0; sg < CLEN / 8; ++sg) {
        if (sg + 2 < CLEN / 8)  // warm L2 ahead (speculative prefetch)
            __builtin_prefetch(xp4 + (size_t)((sg + 2) * 8) * rowstride4, 0, 0);
        float4 xs[8];
#pragma unroll
        for (int j = 0; j < 8; ++j)
            xs[j] = xp4[(size_t)(sg * 8 + j) * rowstride4];
#pragma unroll
        for (int j = 0; j < 8; ++j) {
            const float xv[VEC] = {xs[j].x, xs[j].y, xs[j].z, xs[j].w};
#pragma unroll
            for (int v = 0; v < VEC; ++v)
#pragma unroll
                for (int n = 0; n < NDIM; ++n)
                    h[v][n] = fmaf(q[v][n], h[v][n], xv[v]);
        }
    }
#pragma unroll
    for (int v = 0; v < VEC; ++v) {
        const size_t base = ((size_t)(b * DD + d0 + v) * NDIM) * CHUNKS + c;
#pragma unroll
        for (int n = 0; n < NDIM; ++n) carry[base + (size_t)n * CHUNKS] = h[v][n];
    }
}

// ---------------------------------------------------------------------------
// Kernel 2: serial prefix over chunks (per b,d,n): H_c = E_c + q^CLEN * H_{c-1}
// Overwrites E_c in place with carry-in = H_{c-1}. Tiny (12288 rows x 64).
// ---------------------------------------------------------------------------
__global__ void k2_prefix(const float* __restrict__ delta,
                          const float* __restrict__ alpha,
                          float* __restrict__ carry) {
    const int idx = blockIdx.x * blockDim.x + threadIdx.x;  // (b*DD+d)*NDIM + n
    if (idx >= BB * DD * NDIM) return;
    const int n = idx % NDIM;
    const int d = (idx / NDIM) % DD;

    float p  = sigmoid_f(delta[d * NDIM + n]);
    float qv = 1.0f - p * sigmoid_f(alpha[d * NDIM + n]);
    float qL = __powf(qv, (float)CLEN);

    float* row = carry + (size_t)idx * CHUNKS;
    float H = 0.0f;
    for (int c = 0; c < CHUNKS; ++c) {
        float e = row[c];
        row[c]  = H;                 // carry-in for chunk c
        H = fmaf(qL, H, e);
    }
}

// ---------------------------------------------------------------------------
// Kernel 3: final scan seeded with carry-in, fused silu(y + x*omega) epilogue.
// x tiles staged to LDS via async global->LDS DMA (double buffered). Async
// instruction k copies tile row k; lane l moves one float4, so each wave moves
// a contiguous 512B row-segment and copies exactly the columns it consumes
// (wave-private): s_wait_asynccnt is the only synchronization needed.
// ---------------------------------------------------------------------------
#if USE_ASYNC_LDS
#define ISSUE_TILE(t, buf)                                                      \
    {                                                                           \
        _Pragma("unroll")                                                       \
        for (int k = 0; k < STEP; ++k) {                                        \
            const float* gsrc = xbase + (size_t)((t) * STEP + k) * DD + col;    \
            float* ldst = &tile[(buf)][k * DD + col];                           \
            __builtin_amdgcn_global_load_async_to_lds_b128(                     \
                (g_v4i_ptr)(uintptr_t)gsrc,                                     \
                (l_v4i_ptr)(uint32_t)(uintptr_t)ldst,                           \
                0, 0);                                                          \
        }                                                                       \
    }
#endif

__global__ void __launch_bounds__(NT)
k3_scan_out(const float* __restrict__ x, const float* __restrict__ delta,
            const float* __restrict__ alpha, const float* __restrict__ beta,
            const float* __restrict__ gamma, const float* __restrict__ carry,
            const float* __restrict__ omega, float* __restrict__ out) {
    const int c   = blockIdx.x;
    const int b   = blockIdx.z;
    const int tid = threadIdx.x;
    const int d0  = tid * VEC;

    float q[VEC][NDIM], w[VEC][NDIM];
    load_qw(delta, alpha, beta, gamma, d0, q, w);
    float om[VEC];
#pragma unroll
    for (int v = 0; v < VEC; ++v) om[v] = omega[d0 + v];

    // Seed states with global carry-in: h[c*CLEN - 1]
    float h[VEC][NDIM];
#pragma unroll
    for (int v = 0; v < VEC; ++v) {
        const size_t base = ((size_t)(b * DD + d0 + v) * NDIM) * CHUNKS + c;
#pragma unroll
        for (int n = 0; n < NDIM; ++n) h[v][n] = carry[base + (size_t)n * CHUNKS];
    }

    const float* xbase = x + ((size_t)b * SS + (size_t)c * CLEN) * DD;
    float4* obase4 =
        (float4*)(out + ((size_t)b * SS + (size_t)c * CLEN) * DD) + tid;
    const int rowstride4 = DD / 4;

#if USE_ASYNC_LDS
    __shared__ float tile[2][STEP * DD];  // 2 x 32KB double buffer
    const int col = d0;  // lane's float4 column == its own channels

    ISSUE_TILE(0, 0);
    ISSUE_TILE(1, 1);

    for (int t = 0; t < TILES; ++t) {
        // Async loads complete in order: <=STEP outstanding -> tile t in LDS.
        if (t + 2 <= TILES) __builtin_amdgcn_s_wait_asynccnt(STEP);
        else                __builtin_amdgcn_s_wait_asynccnt(0);

        const float4* lt4 = (const float4*)&tile[t & 1][0] + tid;
#pragma unroll
        for (int s = 0; s < STEP; ++s) {
            float4 xq = lt4[s * rowstride4];       // ds_load_b128
            const float xv[VEC] = {xq.x, xq.y, xq.z, xq.w};
            float o[VEC];
#pragma unroll
            for (int v = 0; v < VEC; ++v) {
#pragma unroll
                for (int n = 0; n < NDIM; ++n)
                    h[v][n] = fmaf(q[v][n], h[v][n], xv[v]);
                float y = fmaf(w[v][0], h[v][0],
                               fmaf(w[v][1], h[v][1], w[v][2] * h[v][2]));
                float z = fmaf(xv[v], om[v], y);
                o[v]    = z * sigmoid_f(z);
            }
            obase4[(size_t)(t * STEP + s) * rowstride4] =
                make_float4(o[0], o[1], o[2], o[3]);
        }
        if (t + 2 < TILES) ISSUE_TILE(t + 2, t & 1);
    }
#else
    // Fallback: direct coalesced float4 loads (x is L2-resident after k1).
    const float4* xb4 = (const float4*)xbase + tid;
    for (int sg = 0; sg < CLEN / 8; ++sg) {
        float4 xs[8];
#pragma unroll
        for (int j = 0; j < 8; ++j)
            xs[j] = xb4[(size_t)(sg * 8 + j) * rowstride4];
#pragma unroll
        for (int j = 0; j < 8; ++j) {
            const float xv[VEC] = {xs[j].x, xs[j].y, xs[j].z, xs[j].w};
            float o[VEC];
#pragma unroll
            for (int v = 0; v < VEC; ++v) {
#pragma unroll
                for (int n = 0; n < NDIM; ++n)
                    h[v][n] = fmaf(q[v][n], h[v][n], xv[v]);
                float y = fmaf(w[v][0], h[v][0],
                               fmaf(w[v][1], h[v][1], w[v][2] * h[v][2]));
                float z = fmaf(xv[v], om[v], y);
                o[v]    = z * sigmoid_f(z);
            }
            obase4[(size_t)(sg * 8 + j) * rowstride4] =
                make_float4(o[0], o[1], o[2], o[3]);
        }
    }
#endif
}

// ---------------------------------------------------------------------------
extern "C" void kernel_launch(void* const* d_in, const int* in_sizes, int n_in,
                              void* d_out, int out_size, void* d_ws, size_t ws_size,
                              hipStream_t stream) {
    const float* x     = (const float*)d_in[0];
    const float* delta = (const float*)d_in[1];
    const float* alpha = (const float*)d_in[2];
    const float* beta  = (const float*)d_in[3];
    const float* gamma = (const float*)d_in[4];
    const float* omega = (const float*)d_in[5];
    float*       out   = (float*)d_out;
    float*       carry = (float*)d_ws;  // B*D*NDIM*CHUNKS floats = 3 MB

    dim3 grid(CHUNKS, 1, BB);

    k1_chunk_end<<<grid, NT, 0, stream>>>(x, delta, alpha, carry);

    const int rows = BB * DD * NDIM;
    k2_prefix<<<(rows + 255) / 256, 256, 0, stream>>>(delta, alpha, carry);

    k3_scan_out<<<grid, NT, 0, stream>>>(x, delta, alpha, beta, gamma, carry,
                                         omega, out);
}